// MemConvAE_20289425506969
// MI455X (gfx1250) — compile-verified
//
#include <hip/hip_runtime.h>
#include <hip/hip_bf16.h>

typedef __attribute__((ext_vector_type(2))) float v2f;
typedef __attribute__((ext_vector_type(8))) float v8f;

#define BATCH     32768
#define LEN       1024
#define FEA       1028
#define MEMD      50
#define DATED     32
#define KPAD      52          // 13 * 4
#define KSTEPS    13
#define NPAD      1040        // 65 * 16, covers FEA + last-tile slack
#define SHRINK_C  0.0025f
#define EPS_C     1e-12f
#define STRIP_OUT 512         // output cols per strip
#define STRIP_TILES 33        // 33*16 = 528 >= 512 + 4 halo
#define STRIP_COLS (STRIP_TILES * 16)

// ---- fp32 WMMA wrapper (D = A(16x4) * B(4x16) + C(16x16)) -----------------
__device__ inline v8f wmma_f32_16x16x4(v2f a, v2f b, v8f c) {
#if defined(__HIP_DEVICE_COMPILE__)
    return __builtin_amdgcn_wmma_f32_16x16x4_f32(
        /*neg_a=*/false, a, /*neg_b=*/false, b,
        /*c_mod=*/(short)0, c, /*reuse_a=*/false, /*reuse_b=*/false);
#else
    (void)a; (void)b;
    return c;
#endif
}

// ---- Stage 0: zero-padded mem_W copy (52 x 1040) so the GEMM loop is ------
// branch-free (no exec-mask churn around v_wmma).
__global__ __launch_bounds__(256) void padW_kernel(
    const float* __restrict__ memW,   // 50 x 1028
    float* __restrict__ memWp)        // 52 x 1040
{
    const int i = blockIdx.x * 256 + threadIdx.x;
    if (i >= KPAD * NPAD) return;
    const int k = i / NPAD, n = i % NPAD;
    memWp[i] = (k < MEMD && n < FEA) ? memW[(size_t)k * FEA + n] : 0.f;
}

// ---- Stage 1a: attention rows ---------------------------------------------
// att[row, m] for m<50, zero-padded to KPAD so WMMA A needs no K guard.
__global__ __launch_bounds__(256) void att_kernel(
    const float* __restrict__ dv,     // B x 32
    const float* __restrict__ dateW,  // 50 x 32
    float* __restrict__ attp)         // B x KPAD (workspace)
{
    __shared__ float dw_s[MEMD * DATED];
    const int tid = threadIdx.x;
    for (int i = tid; i < MEMD * DATED; i += 256) dw_s[i] = dateW[i];
    __syncthreads();

    const int row = blockIdx.x * 256 + tid;

    float v[DATED];
    const float4* p = (const float4*)(dv + (size_t)row * DATED);
#pragma unroll
    for (int j = 0; j < DATED / 4; ++j) {
        float4 q = p[j];
        v[4*j+0] = q.x; v[4*j+1] = q.y; v[4*j+2] = q.z; v[4*j+3] = q.w;
    }

    float sc[MEMD];
    float mx = -1e30f;
#pragma unroll 1
    for (int m = 0; m < MEMD; ++m) {
        float a = 0.f;
#pragma unroll
        for (int k = 0; k < DATED; ++k) a = fmaf(v[k], dw_s[m * DATED + k], a);
        sc[m] = a;
        mx = fmaxf(mx, a);
    }
    float sum = 0.f;
#pragma unroll 1
    for (int m = 0; m < MEMD; ++m) { float e = __expf(sc[m] - mx); sc[m] = e; sum += e; }
    const float inv = 1.f / sum;
    float l1 = 0.f;
#pragma unroll 1
    for (int m = 0; m < MEMD; ++m) {
        float a = sc[m] * inv;
        float d = a - SHRINK_C;
        float t = fmaxf(d, 0.f) * a / (fabsf(d) + EPS_C);   // soft-shrink gate
        sc[m] = t;
        l1 += fabsf(t);
    }
    const float invl1 = 1.f / (l1 + EPS_C);
    float* o = attp + (size_t)row * KPAD;
#pragma unroll 1
    for (int m = 0; m < MEMD; ++m) o[m] = sc[m] * invl1;
    o[MEMD] = 0.f; o[MEMD + 1] = 0.f;                       // K padding
}

// ---- Stage 1b: per-row weight / bias (reads x: the big 134MB stream) ------
__global__ __launch_bounds__(256) void wb_kernel(
    const float* __restrict__ x,      // B x L
    const float* __restrict__ fcw_w,  // L
    const float* __restrict__ fcw_b,  // 1
    const float* __restrict__ fcb_w,  // L
    const float* __restrict__ fcb_b,  // 1
    float* __restrict__ wout,         // B
    float* __restrict__ bout)         // B
{
    const int wave = threadIdx.x >> 5;
    const int lane = threadIdx.x & 31;
    const int row  = blockIdx.x * 8 + wave;

    const float4* xr = (const float4*)(x + (size_t)row * LEN);
    const float4* ww = (const float4*)fcw_w;
    const float4* bw = (const float4*)fcb_w;

    float aw = 0.f, ab = 0.f;
#pragma unroll
    for (int j = 0; j < LEN / (32 * 4); ++j) {
        const int idx = lane + 32 * j;                       // coalesced b128
        float4 xv = xr[idx];
        float4 wv = ww[idx];
        float4 bv = bw[idx];
        aw = fmaf(xv.x, wv.x, fmaf(xv.y, wv.y, fmaf(xv.z, wv.z, fmaf(xv.w, wv.w, aw))));
        ab = fmaf(xv.x, bv.x, fmaf(xv.y, bv.y, fmaf(xv.z, bv.z, fmaf(xv.w, bv.w, ab))));
    }
#pragma unroll
    for (int off = 16; off > 0; off >>= 1) {
        aw += __shfl_xor(aw, off, 32);
        ab += __shfl_xor(ab, off, 32);
    }
    if (lane == 0) {
        wout[row] = tanhf(aw + fcw_b[0]) * 0.5f + 1.0f;
        bout[row] = tanhf(ab + fcb_b[0]) * 0.5f;
    }
}

// ---- Stage 2: WMMA GEMM (att @ mem_W) + 5-tap FIR + affine ----------------
__global__ __launch_bounds__(256) void gemm_conv_kernel(
    const float* __restrict__ attp,   // B x KPAD (zero-padded K)
    const float* __restrict__ memWp,  // KPAD x NPAD (zero-padded)
    const float* __restrict__ convt_w,// 5
    const float* __restrict__ wrow,   // B
    const float* __restrict__ brow,   // B
    float* __restrict__ out)          // B x L
{
    __shared__ float att_s[16][KPAD];            // 3.3 KB
    __shared__ float st_s[16][STRIP_COLS + 4];   // 34.0 KB

    const int m0   = blockIdx.x * 16;
    const int tid  = threadIdx.x;
    const int wave = tid >> 5;
    const int lane = tid & 31;
    const int l16  = lane & 15;
    const int khalf = (lane >> 4) * 2;           // lanes 16-31 hold K+2,K+3
    const int rbase = (lane >> 4) * 8;           // C rows: v + 8*(lane>=16)

    // stage att tile (16 x KPAD) into LDS
    for (int i = tid; i < 16 * KPAD; i += 256) {
        const int r = i / KPAD, k = i % KPAD;
        att_s[r][k] = attp[(size_t)(m0 + r) * KPAD + k];
    }
    __syncthreads();

    // pre-stage all 13 A-fragments in VGPRs (16x4 f32 layout)
    v2f afrag[KSTEPS];
#pragma unroll
    for (int ks = 0; ks < KSTEPS; ++ks) {
        afrag[ks].x = att_s[l16][ks * 4 + khalf + 0];
        afrag[ks].y = att_s[l16][ks * 4 + khalf + 1];
    }

    // flipped conv taps: out[i] = sum_k st[i+k] * convt_w[4-k]
    float cw[5];
#pragma unroll
    for (int k = 0; k < 5; ++k) cw[k] = convt_w[4 - k];

    const int r_out = tid >> 4;                  // conv phase: row 0..15
    const int c16   = tid & 15;
    const float wr = wrow[m0 + r_out];
    const float br = brow[m0 + r_out];

    // lane-dependent part of the B pointer folded in once: khalf row
    const float* bbase = memWp + (size_t)khalf * NPAD;

#pragma unroll 1
    for (int s = 0; s < 2; ++s) {
        const int n0s = s * STRIP_OUT;           // strip base st_mem column

        // --- WMMA tile compute: 33 N-tiles split over 8 waves, branch-free ---
        for (int t = wave; t < STRIP_TILES; t += 8) {
            const int n = n0s + t * 16 + l16;    // B-fragment column
            const float* bcol = bbase + n;       // uniform imm offsets below
            v8f c = {};
#pragma unroll
            for (int ks = 0; ks < KSTEPS; ++ks) {
                v2f b;
                b.x = bcol[ks * 4 * NPAD];           // row 4ks+khalf
                b.y = bcol[ks * 4 * NPAD + NPAD];    // row 4ks+khalf+1
                c = wmma_f32_16x16x4(afrag[ks], b, c);
            }
            const int lc = t * 16 + l16;
#pragma unroll
            for (int v = 0; v < 8; ++v) st_s[rbase + v][lc] = c[v];
        }
        __syncthreads();

        // --- 5-tap FIR + per-row affine, coalesced store ---
#pragma unroll 4
        for (int j = 0; j < STRIP_OUT / 16; ++j) {
            const int c = c16 + j * 16;          // 0..511, lane-strided
            float acc = st_s[r_out][c + 0] * cw[0];
            acc = fmaf(st_s[r_out][c + 1], cw[1], acc);
            acc = fmaf(st_s[r_out][c + 2], cw[2], acc);
            acc = fmaf(st_s[r_out][c + 3], cw[3], acc);
            acc = fmaf(st_s[r_out][c + 4], cw[4], acc);
            out[(size_t)(m0 + r_out) * LEN + n0s + c] = fmaf(acc, wr, br);
        }
        __syncthreads();                         // st_s reused next strip
    }
}

// ---------------------------------------------------------------------------
extern "C" void kernel_launch(void* const* d_in, const int* in_sizes, int n_in,
                              void* d_out, int out_size, void* d_ws, size_t ws_size,
                              hipStream_t stream) {
    (void)in_sizes; (void)n_in; (void)out_size; (void)ws_size;
    const float* x           = (const float*)d_in[0];
    const float* date_vector = (const float*)d_in[1];
    // d_in[2] (conv_w) is dead code in the reference: its conv output is
    // overwritten by st_mem before use.
    const float* convt_w     = (const float*)d_in[3];
    const float* mem_W       = (const float*)d_in[4];
    const float* date_W      = (const float*)d_in[5];
    const float* fcw_w       = (const float*)d_in[6];
    const float* fcw_b       = (const float*)d_in[7];
    const float* fcb_w       = (const float*)d_in[8];
    const float* fcb_b       = (const float*)d_in[9];

    float* out  = (float*)d_out;                 // [st | weight | bias]
    float* st   = out;
    float* wout = out + (size_t)BATCH * LEN;
    float* bout = wout + BATCH;

    float* attp  = (float*)d_ws;                         // B x KPAD
    float* memWp = attp + (size_t)BATCH * KPAD;          // KPAD x NPAD

    padW_kernel<<<(KPAD * NPAD + 255) / 256, 256, 0, stream>>>(mem_W, memWp);
    att_kernel<<<BATCH / 256, 256, 0, stream>>>(date_vector, date_W, attp);
    wb_kernel<<<BATCH / 8, 256, 0, stream>>>(x, fcw_w, fcw_b, fcb_w, fcb_b, wout, bout);
    gemm_conv_kernel<<<BATCH / 16, 256, 0, stream>>>(attp, memWp, convt_w, wout, bout, st);
}